// DeepAttention_25786983645960
// MI455X (gfx1250) — compile-verified
//
#include <hip/hip_runtime.h>
#include <math.h>
#include <stdint.h>

// Problem dims (fixed by the reference)
#define B_N 128
#define S_N 8192
#define D_N 128
#define NEG_INF (-1e12f)

typedef __attribute__((ext_vector_type(2))) float v2f;
typedef __attribute__((ext_vector_type(8))) float v8f;
typedef __attribute__((ext_vector_type(4))) unsigned int u32x4;
typedef __attribute__((ext_vector_type(8))) int i32x8;
typedef __attribute__((ext_vector_type(4))) int i32x4;

#if __has_builtin(__builtin_amdgcn_tensor_load_to_lds) && \
    __has_builtin(__builtin_amdgcn_s_wait_tensorcnt)
#define USE_TDM 1
#else
#define USE_TDM 0
#endif

// ---------------- Kernel 1: g[b,e] = relu(input[b]·W[e]) * V[e] ----------------
__global__ void k1_g(const float* __restrict__ inp, const float* __restrict__ W,
                     const float* __restrict__ V, float* __restrict__ g) {
  const int b = blockIdx.x, e = threadIdx.x;
  const float* x = inp + b * D_N;
  const float* w = W + e * D_N;
  float acc = 0.f;
#pragma unroll 4
  for (int d = 0; d < D_N; ++d) acc += x[d] * w[d];
  g[b * D_N + e] = fmaxf(acc, 0.f) * V[e];
}

// ---------------- Kernel 2: logits[b,s] = g[b] · relu(context[b,s]·Wᵀ) --------
// WMMA f32 16x16x4.  LDS rows padded to stride 132 dwords so (row,k0..k0+3)
// -> bank 4*row+k is bijective over 64 banks: conflict-free ds_load_b64
// fragment fetches.  Context tiles are DMA'd by the Tensor Data Mover with
// hardware padding (4 dwords every 128 dwords), double-buffered per wave,
// synchronized only with s_wait_tensorcnt (no block barriers in the s-loop).
#define WPB 8        // waves per block
#define SCHUNK 4     // s-tiles (16 rows) per wave
#define WSTRIDE 132
#define W_FLOATS (128 * WSTRIDE)  // 16896
#define A_FLOATS (16 * WSTRIDE)   // 2112
#define K2_SMEM ((W_FLOATS + 128 + WPB * 2 * A_FLOATS) * sizeof(float))  // ~199KB

#if USE_TDM
// DMA one 16x128 f32 context tile (2048 dwords, contiguous in memory) into LDS
// at byte offset lds_off, inserting 4 pad dwords after every 128 dwords so the
// tile lands with row stride 132.
__device__ __forceinline__ void tdm_load_tile(unsigned lds_off, const float* gsrc) {
  const unsigned long long ga = (unsigned long long)(uintptr_t)gsrc;
  u32x4 g0;
  g0[0] = 1u;                                   // count=1, user mode, no gather
  g0[1] = lds_off;                              // lds_addr (bytes)
  g0[2] = (unsigned)(ga & 0xffffffffu);         // global_addr[31:0]
  g0[3] = (unsigned)((ga >> 32) & 0x01ffffffu)  // global_addr[56:32]
          | (2u << 30);                         // type=2 ("image")
  i32x8 g1;
  g1[0] = (2 << 16)      // data_size = 4 bytes
        | (1 << 20)      // pad_enable
        | (6 << 22)      // pad_interval: every 128 dwords
        | (3 << 25);     // pad_amount: 4 dwords
  g1[1] = (int)(2048u << 16);  // tensor_dim0[15:0]=2048 (bits 63:48)
  g1[2] = (int)(1u << 16);     // tensor_dim0 hi=0; tensor_dim1[15:0]=1
  g1[3] = (int)(2048u << 16);  // tensor_dim1 hi=0; tile_dim0=2048
  g1[4] = 1;                   // tile_dim1=1, tile_dim2=0
  g1[5] = 2048;                // tensor_dim0_stride lo
  g1[6] = 0;
  g1[7] = 0;
  const i32x4 z4 = {0, 0, 0, 0};
  const i32x8 z8 = {0, 0, 0, 0, 0, 0, 0, 0};
  __builtin_amdgcn_tensor_load_to_lds(g0, g1, z4, z4, z8, 0);
}
#endif

__global__ void __launch_bounds__(WPB * 32) k2_logits(
    const float* __restrict__ ctx, const float* __restrict__ W,
    const float* __restrict__ g, float* __restrict__ logits) {
  extern __shared__ float smem[];
  float* Wl = smem;                              // [128][132]
  float* gl = smem + W_FLOATS;                   // [128]
  float* Abase = gl + 128 + (threadIdx.x >> 5) * (2 * A_FLOATS);  // 2 bufs/wave

  const int b = blockIdx.y;
  const int tid = threadIdx.x;
  const int lane = tid & 31;
  const int wave = tid >> 5;
  const int n = lane & 15;   // B col / A row / C col index
  const int hi = lane >> 4;  // lane-half selects K offset 0 vs 2

  // Stage W (row-major, padded) -- coalesced float4 loads.
  for (int i = tid * 4; i < 128 * 128; i += WPB * 32 * 4) {
    const float4 w4 = *(const float4*)(W + i);
    const int r = i >> 7, c = i & 127;
    float* dst = Wl + r * WSTRIDE + c;
    dst[0] = w4.x; dst[1] = w4.y; dst[2] = w4.z; dst[3] = w4.w;
  }
  if (tid < 128) gl[tid] = g[b * D_N + tid];
  __syncthreads();

  const int tile0 = (blockIdx.x * WPB + wave) * SCHUNK;
  const float* src0 = ctx + ((size_t)b * S_N + (size_t)tile0 * 16) * D_N;

#if USE_TDM
  const unsigned lds0 = __builtin_amdgcn_groupstaticsize() +
                        (unsigned)((const char*)Abase - (const char*)smem);
  // Prologue: DMA tile 0 into buffer 0.
  tdm_load_tile(lds0, src0);
#endif

  for (int t = 0; t < SCHUNK; ++t) {
    float* Al = Abase + (t & 1) * A_FLOATS;

#if USE_TDM
    if (t + 1 < SCHUNK) {
      tdm_load_tile(lds0 + (unsigned)(((t + 1) & 1) * A_FLOATS * sizeof(float)),
                    src0 + (size_t)(t + 1) * 16 * D_N);
      __builtin_amdgcn_s_wait_tensorcnt(1);  // tile t landed (in-order)
    } else {
      __builtin_amdgcn_s_wait_tensorcnt(0);
    }
    asm volatile("" ::: "memory");
#else
    // Fallback: stage 16x128 tile with vector loads + barrier.
    const float* src = src0 + (size_t)t * 16 * D_N;
#pragma unroll
    for (int i = 0; i < 16; ++i) {
      const int f4 = lane + i * 32;
      const int r = f4 >> 5;
      const int c4 = (f4 & 31) << 2;
      const float4 v4 = *(const float4*)(src + r * D_N + c4);
      float* dst = Al + r * WSTRIDE + c4;
      dst[0] = v4.x; dst[1] = v4.y; dst[2] = v4.z; dst[3] = v4.w;
    }
    __syncthreads();
#endif

    // Preload all 32 A fragments (reused across the 8 e-tiles).
    // A 16x4 layout: VGPR0 = K0(lo lanes)/K2(hi lanes), VGPR1 = K1/K3.
    v2f a[32];
#pragma unroll
    for (int i = 0; i < 32; ++i)
      a[i] = *(const v2f*)(Al + n * WSTRIDE + i * 4 + 2 * hi);

    float logit[8] = {0.f, 0.f, 0.f, 0.f, 0.f, 0.f, 0.f, 0.f};
#pragma unroll
    for (int et = 0; et < 8; ++et) {
      const float* wb = Wl + (et * 16 + n) * WSTRIDE + 2 * hi;
      v8f c = {};
#pragma unroll
      for (int i = 0; i < 32; ++i) {
        const v2f bb = *(const v2f*)(wb + i * 4);
        c = __builtin_amdgcn_wmma_f32_16x16x4_f32(
            false, a[i], false, bb, (short)0, c, false, false);
      }
      // logit is additive across e-tiles after the relu: += g[e] * relu(v)
      const float gN = gl[et * 16 + n];
#pragma unroll
      for (int j = 0; j < 8; ++j) logit[j] += fmaxf(c[j], 0.f) * gN;
    }

    // Reduce across the 16 columns held by each 16-lane group.
#pragma unroll
    for (int off = 8; off >= 1; off >>= 1) {
#pragma unroll
      for (int j = 0; j < 8; ++j) logit[j] += __shfl_xor(logit[j], off, 32);
    }
    float* lout = logits + (size_t)b * S_N + (size_t)(tile0 + t) * 16;
    if (lane == 0) {
#pragma unroll
      for (int j = 0; j < 8; ++j) lout[j] = logit[j];        // rows 0..7
    } else if (lane == 16) {
#pragma unroll
      for (int j = 0; j < 8; ++j) lout[8 + j] = logit[j];    // rows 8..15
    }
#if !USE_TDM
    __syncthreads();
#endif
  }
}

// ---------------- Kernel 3: masked softmax over S per batch -------------------
__global__ void k3_softmax(const float* __restrict__ logits,
                           const unsigned char* __restrict__ mask,
                           float* __restrict__ attn) {
  const int b = blockIdx.x, tid = threadIdx.x;
  __shared__ float red[256];
  const float* lb = logits + (size_t)b * S_N;
  const unsigned char* mb = mask + (size_t)b * S_N;
  float m = -INFINITY;
  for (int s = tid; s < S_N; s += 256) {
    const float x = mb[s] ? NEG_INF : lb[s];
    m = fmaxf(m, x);
  }
  red[tid] = m;
  __syncthreads();
  for (int off = 128; off; off >>= 1) {
    if (tid < off) red[tid] = fmaxf(red[tid], red[tid + off]);
    __syncthreads();
  }
  const float M = red[0];
  __syncthreads();

  float sum = 0.f;
  float* ab = attn + (size_t)b * S_N;
  for (int s = tid; s < S_N; s += 256) {
    const float x = mb[s] ? NEG_INF : lb[s];
    const float p = expf(x - M);
    ab[s] = p;
    sum += p;
  }
  red[tid] = sum;
  __syncthreads();
  for (int off = 128; off; off >>= 1) {
    if (tid < off) red[tid] += red[tid + off];
    __syncthreads();
  }
  const float inv = 1.f / red[0];
  for (int s = tid; s < S_N; s += 256) ab[s] *= inv;
}

// ---------------- Kernel 4: weighted[b,d] = Σ_s attn[b,s]·context[b,s,d] ------
#define CHUNKS 32
__global__ void k4_weighted_part(const float* __restrict__ ctx,
                                 const float* __restrict__ attn,
                                 float* __restrict__ part) {
  const int b = blockIdx.y, c = blockIdx.x, d = threadIdx.x;
  const int s0 = c * (S_N / CHUNKS);
  const float* ab = attn + (size_t)b * S_N + s0;
  const float* cb = ctx + ((size_t)b * S_N + s0) * D_N + d;
  float acc = 0.f;
#pragma unroll 4
  for (int s = 0; s < S_N / CHUNKS; ++s) acc += ab[s] * cb[(size_t)s * D_N];
  part[((size_t)b * CHUNKS + c) * D_N + d] = acc;
}

__global__ void k4b_reduce(const float* __restrict__ part, float* __restrict__ wsum) {
  const int b = blockIdx.x, d = threadIdx.x;
  float acc = 0.f;
#pragma unroll
  for (int c = 0; c < CHUNKS; ++c) acc += part[((size_t)b * CHUNKS + c) * D_N + d];
  wsum[b * D_N + d] = acc;
}

// ---------------- Kernel 5: h = tanh(concat(weighted,input)·Woutᵀ) ------------
__global__ void k5_out(const float* __restrict__ wsum, const float* __restrict__ inp,
                       const float* __restrict__ Wout, float* __restrict__ out) {
  const int b = blockIdx.x, e = threadIdx.x;
  const float* wo = Wout + e * (2 * D_N);
  const float* wb = wsum + b * D_N;
  const float* xb = inp + b * D_N;
  float acc = 0.f;
#pragma unroll 4
  for (int f = 0; f < D_N; ++f) acc += wb[f] * wo[f] + xb[f] * wo[D_N + f];
  out[b * D_N + e] = tanhf(acc);
}

// ---------------- Launcher ----------------------------------------------------
extern "C" void kernel_launch(void* const* d_in, const int* in_sizes, int n_in,
                              void* d_out, int out_size, void* d_ws, size_t ws_size,
                              hipStream_t stream) {
  const float* inp = (const float*)d_in[0];                   // [B, D]
  const float* ctx = (const float*)d_in[1];                   // [B, S, D]
  const unsigned char* mask = (const unsigned char*)d_in[2];  // [B, S] bool
  const float* W = (const float*)d_in[3];                     // [D, D]
  const float* V = (const float*)d_in[4];                     // [1, D]
  const float* Wout = (const float*)d_in[5];                  // [D, 2D]

  float* out = (float*)d_out;
  float* h = out;                      // [B, D]  (first output)
  float* attn = out + B_N * D_N;       // [B, S]  (second output)

  float* ws = (float*)d_ws;
  float* g = ws;                                       // [B, D]
  float* logits = ws + B_N * D_N;                      // [B, S]
  float* part = logits;                                // reuse after softmax
  float* wsum = ws + B_N * D_N + (size_t)B_N * S_N;    // [B, D]

  (void)in_sizes; (void)n_in; (void)out_size; (void)ws_size;

  (void)hipFuncSetAttribute(reinterpret_cast<const void*>(k2_logits),
                            hipFuncAttributeMaxDynamicSharedMemorySize,
                            (int)K2_SMEM);

  k1_g<<<B_N, D_N, 0, stream>>>(inp, W, V, g);
  k2_logits<<<dim3(S_N / (16 * SCHUNK * WPB), B_N), WPB * 32, K2_SMEM, stream>>>(
      ctx, W, g, logits);
  k3_softmax<<<B_N, 256, 0, stream>>>(logits, mask, attn);
  k4_weighted_part<<<dim3(CHUNKS, B_N), D_N, 0, stream>>>(ctx, attn, part);
  k4b_reduce<<<B_N, D_N, 0, stream>>>(part, wsum);
  k5_out<<<B_N, D_N, 0, stream>>>(wsum, inp, Wout, h);
}